// LinearContentDecoder_6536940224703
// MI455X (gfx1250) — compile-verified
//
#include <hip/hip_runtime.h>

typedef __attribute__((ext_vector_type(2))) float v2f;
typedef __attribute__((ext_vector_type(8))) float v8f;

#define NPRIMS  4096
#define INCH    256
#define FIRSTCH 256
#define NBASES  32
#define BATCH   32
#define PRIMOUT 2048

// D = A(16x4 f32) * B(4x16 f32) + C(16x16 f32), wave32 matrix op.
__device__ __forceinline__ v8f wmma_f32(v2f a, v2f b, v8f c) {
  return __builtin_amdgcn_wmma_f32_16x16x4_f32(
      /*neg_a=*/false, a, /*neg_b=*/false, b,
      /*c_mod=*/(short)0, c, /*reuse_a=*/false, /*reuse_b=*/false);
}

// ---------------------------------------------------------------------------
// Kernel A: h = leaky_relu(x @ W0 + b0, 0.2)   [32 x 256]  (trivial FLOPs)
// grid = 32 (batch), block = 256 (out channel)
// ---------------------------------------------------------------------------
__global__ void __launch_bounds__(256) dec0_kernel(
    const float* __restrict__ x, const float* __restrict__ W0,
    const float* __restrict__ b0, float* __restrict__ h) {
  const int b = blockIdx.x;   // batch row
  const int j = threadIdx.x;  // output channel
  const float* xr = x + b * INCH;
  float acc = b0[j];
#pragma unroll 8
  for (int c = 0; c < INCH; ++c)
    acc = fmaf(xr[c], W0[c * FIRSTCH + j], acc);  // W0 row-major: coalesced in j
  h[b * FIRSTCH + j] = acc > 0.0f ? acc : 0.2f * acc;
}

// ---------------------------------------------------------------------------
// Kernel B: z[n] = h @ W1[n] + b1[n]   (M=32, K=256, N=32 per primitive)
// grid = 512 blocks x 256 threads; one wave per primitive; h staged in LDS.
// ---------------------------------------------------------------------------
__global__ void __launch_bounds__(256) bottleneck_kernel(
    const float* __restrict__ h, const float* __restrict__ W1,
    const float* __restrict__ b1, float* __restrict__ z) {
  __shared__ float lds_h[BATCH * FIRSTCH];  // 32 KB
  for (int i = threadIdx.x; i < BATCH * FIRSTCH; i += 256)
    lds_h[i] = h[i];
  __syncthreads();

  const int wave = threadIdx.x >> 5;
  const int lane = threadIdx.x & 31;
  const int l16  = lane & 15;
  const int hi   = lane >> 4;                 // lane half selects K+2 (A/B) & M+8 (C)
  const int n    = blockIdx.x * 8 + wave;     // primitive id

  const float* Wn = W1 + (size_t)n * FIRSTCH * NBASES;

  // C init = bias broadcast along M (bias depends only on N = lane column)
  v8f c[2][2];
#pragma unroll
  for (int nt = 0; nt < 2; ++nt) {
    const float bias = b1[n * NBASES + nt * 16 + l16];
    v8f cv;
#pragma unroll
    for (int r = 0; r < 8; ++r) cv[r] = bias;
    c[0][nt] = cv;
    c[1][nt] = cv;
  }

  for (int ks = 0; ks < FIRSTCH; ks += 4) {
    const int kk = ks + 2 * hi;               // this lane-half's K pair
    v2f a[2], b[2];
#pragma unroll
    for (int m2 = 0; m2 < 2; ++m2)            // A: h[M][kk], h[M][kk+1] (8B LDS read)
      a[m2] = *(const v2f*)&lds_h[(m2 * 16 + l16) * FIRSTCH + kk];
#pragma unroll
    for (int nt = 0; nt < 2; ++nt) {          // B: W1[n][kk][col], row stride 32
      const float* p = Wn + (size_t)kk * NBASES + nt * 16 + l16;
      v2f bv;
      bv.x = p[0];
      bv.y = p[NBASES];
      b[nt] = bv;
    }
#pragma unroll
    for (int m2 = 0; m2 < 2; ++m2)
#pragma unroll
      for (int nt = 0; nt < 2; ++nt)
        c[m2][nt] = wmma_f32(a[m2], b[nt], c[m2][nt]);
  }

  // Store z[n] as [b(M)=32][d(N)=32]
  float* zn = z + (size_t)n * BATCH * NBASES;
#pragma unroll
  for (int m2 = 0; m2 < 2; ++m2)
#pragma unroll
    for (int nt = 0; nt < 2; ++nt)
#pragma unroll
      for (int r = 0; r < 8; ++r) {
        const int M = m2 * 16 + r + 8 * hi;   // C/D layout: VGPR r -> M=r / M=r+8
        const int N = nt * 16 + l16;
        zn[M * NBASES + N] = c[m2][nt][r];
      }
}

// ---------------------------------------------------------------------------
// Kernel C: y[n] = z[n] @ W2[n] + b2[n]  (M=32, K=32, N=2048 per primitive)
// This streams the 1 GB of W2 and writes the 1 GB output -> bandwidth kernel.
// grid = (4096, 8): block = (primitive, 256-column chunk); 8 waves x 32 cols.
// ---------------------------------------------------------------------------
__global__ void __launch_bounds__(256) expand_kernel(
    const float* __restrict__ z, const float* __restrict__ W2,
    const float* __restrict__ b2, float* __restrict__ out) {
  __shared__ float lds_z[BATCH * NBASES];  // 4 KB
  const int n = blockIdx.x;
  for (int i = threadIdx.x; i < BATCH * NBASES; i += 256)
    lds_z[i] = z[(size_t)n * BATCH * NBASES + i];
  __syncthreads();

  const int wave = threadIdx.x >> 5;
  const int lane = threadIdx.x & 31;
  const int l16  = lane & 15;
  const int hi   = lane >> 4;
  const int nb   = blockIdx.y * 256 + wave * 32;  // this wave's column base

  // Preload entire A (z tile, both M halves, all 8 k-steps): 32 VGPRs, reused
  // across every N tile this wave produces.
  v2f a[2][8];
#pragma unroll
  for (int m2 = 0; m2 < 2; ++m2)
#pragma unroll
    for (int ks8 = 0; ks8 < 8; ++ks8) {
      const int kk = ks8 * 4 + 2 * hi;
      a[m2][ks8] = *(const v2f*)&lds_z[(m2 * 16 + l16) * NBASES + kk];
    }

  const float* Wn = W2 + (size_t)n * NBASES * PRIMOUT;

  v8f c[2][2];
#pragma unroll
  for (int nt = 0; nt < 2; ++nt) {
    const float bias = b2[(size_t)n * PRIMOUT + nb + nt * 16 + l16];
    v8f cv;
#pragma unroll
    for (int r = 0; r < 8; ++r) cv[r] = bias;
    c[0][nt] = cv;
    c[1][nt] = cv;
  }

#pragma unroll
  for (int ks8 = 0; ks8 < 8; ++ks8) {
    const int kk = ks8 * 4 + 2 * hi;
    v2f b[2];
#pragma unroll
    for (int nt = 0; nt < 2; ++nt) {          // B: W2[n][kk][col], row stride 2048
      const float* p = Wn + (size_t)kk * PRIMOUT + nb + nt * 16 + l16;
      v2f bv;
      bv.x = p[0];
      bv.y = p[PRIMOUT];
      b[nt] = bv;
    }
#pragma unroll
    for (int m2 = 0; m2 < 2; ++m2)
#pragma unroll
      for (int nt = 0; nt < 2; ++nt)
        c[m2][nt] = wmma_f32(a[m2][ks8], b[nt], c[m2][nt]);
  }

  // Store: y[b][n][e] flat = ((b*4096)+n)*2048 + e ; per-VGPR b32 stores are
  // two 64B coalesced segments (lane halves are M and M+8 rows).
#pragma unroll
  for (int m2 = 0; m2 < 2; ++m2)
#pragma unroll
    for (int nt = 0; nt < 2; ++nt)
#pragma unroll
      for (int r = 0; r < 8; ++r) {
        const int M = m2 * 16 + r + 8 * hi;   // batch index
        const int N = nb + nt * 16 + l16;     // output column
        out[((size_t)M * NPRIMS + n) * PRIMOUT + N] = c[m2][nt][r];
      }
}

// ---------------------------------------------------------------------------
extern "C" void kernel_launch(void* const* d_in, const int* in_sizes, int n_in,
                              void* d_out, int out_size, void* d_ws, size_t ws_size,
                              hipStream_t stream) {
  const float* x  = (const float*)d_in[0];
  const float* W0 = (const float*)d_in[1];
  const float* b0 = (const float*)d_in[2];
  const float* W1 = (const float*)d_in[3];
  const float* b1 = (const float*)d_in[4];
  const float* W2 = (const float*)d_in[5];
  const float* b2 = (const float*)d_in[6];
  float* out = (float*)d_out;

  float* h = (float*)d_ws;                        // 32*256 f32  = 32 KB
  float* z = (float*)((char*)d_ws + 32 * 1024);   // 4096*32*32 f32 = 16 MB

  dec0_kernel<<<dim3(BATCH), dim3(FIRSTCH), 0, stream>>>(x, W0, b0, h);
  bottleneck_kernel<<<dim3(NPRIMS / 8), dim3(256), 0, stream>>>(h, W1, b1, z);
  expand_kernel<<<dim3(NPRIMS, PRIMOUT / 256), dim3(256), 0, stream>>>(z, W2, b2, out);
}